// MultiHeadAttention_80642305950243
// MI455X (gfx1250) — compile-verified
//
#include <hip/hip_runtime.h>

typedef __attribute__((ext_vector_type(2))) float v2f;
typedef __attribute__((ext_vector_type(8))) float v8f;

#define EMB   1024
#define HEADS 16
#define DH    64
#define BATCH 4
#define SEQ   2048
#define NTOK  (BATCH * SEQ)

// LDS byte address = low 32 bits of the generic (flat) shared-pointer value.
__device__ __forceinline__ unsigned lds_addr_of(const void* p) {
  return (unsigned)(uintptr_t)p;
}

// CDNA5 async global->LDS copies (ASYNCcnt engine, per-lane LDS addressing).
__device__ __forceinline__ void async_g2l_b32(unsigned lds, unsigned goff,
                                              unsigned long long base) {
  asm volatile("global_load_async_to_lds_b32 %0, %1, %2"
               :: "v"(lds), "v"(goff), "s"(base) : "memory");
}
__device__ __forceinline__ void async_g2l_b64(unsigned lds, unsigned goff,
                                              unsigned long long base) {
  asm volatile("global_load_async_to_lds_b64 %0, %1, %2"
               :: "v"(lds), "v"(goff), "s"(base) : "memory");
}
__device__ __forceinline__ void async_g2l_b128(unsigned lds, unsigned goff,
                                               unsigned long long base) {
  asm volatile("global_load_async_to_lds_b128 %0, %1, %2"
               :: "v"(lds), "v"(goff), "s"(base) : "memory");
}
__device__ __forceinline__ void wait_async0() {
  asm volatile("s_wait_asynccnt 0x0" ::: "memory");
}

// ---------------------------------------------------------------------------
// GEMM: out[M,N] = A[M,EMB] @ W[EMB,EMB] + bias ;  M = NTOK, K = N = EMB
// Block tile 128x128, 8 waves (4x2), each wave a 32x64 sub-tile (2x4 C tiles).
// Tiles staged with GLOBAL_LOAD_ASYNC_TO_LDS (B transposed in-flight).
// ---------------------------------------------------------------------------
#define BM 128
#define BN 128
#define BK 16
#define LDA 18   // As row stride (BK+2): float2-aligned, conflict-free
#define LDB 18   // Bst row stride

__global__ __launch_bounds__(256) void gemm_bias_kernel(const float* __restrict__ A,
                                                        const float* __restrict__ W,
                                                        const float* __restrict__ bias,
                                                        float* __restrict__ out) {
  __shared__ float As[BM * LDA];     // [row][k]
  __shared__ float Bst[BN * LDB];    // [col][k]  (transposed)
  const int t    = threadIdx.x;
  const int lane = t & 31;
  const int w    = t >> 5;
  const int half = lane >> 4;
  const int l16  = lane & 15;
  const int wr   = w >> 1;           // 0..3 -> 32-row strip
  const int wc   = w & 1;            // 0..1 -> 64-col strip
  const int m0   = blockIdx.y * BM;
  const int n0   = blockIdx.x * BN;
  const int rt0  = wr * 32;
  const int ct0  = wc * 64;

  const unsigned long long Abase = (unsigned long long)(uintptr_t)A;
  const unsigned long long Wbase = (unsigned long long)(uintptr_t)W;

  v8f zero = {};
  v8f acc[2][4];
#pragma unroll
  for (int mi = 0; mi < 2; ++mi)
#pragma unroll
    for (int ni = 0; ni < 4; ++ni) acc[mi][ni] = zero;

  for (int k0 = 0; k0 < EMB; k0 += BK) {
    __syncthreads();                 // previous-iter LDS reads complete
    // A tile: 128x16 floats as async b64 (contiguous in both spaces)
#pragma unroll
    for (int j = 0; j < 4; ++j) {
      int i   = j * 256 + t;
      int row = i >> 3;
      int c2  = (i & 7) << 1;
      async_g2l_b64(lds_addr_of(&As[row * LDA + c2]),
                    (unsigned)(((m0 + row) * EMB + k0 + c2) * 4), Abase);
    }
    // B tile: 16x128 floats, transposed into Bst via per-lane LDS addresses
#pragma unroll
    for (int j = 0; j < 8; ++j) {
      int i  = j * 256 + t;
      int kk = i >> 7;
      int n  = i & 127;
      async_g2l_b32(lds_addr_of(&Bst[n * LDB + kk]),
                    (unsigned)(((k0 + kk) * EMB + n0 + n) * 4), Wbase);
    }
    wait_async0();
    __syncthreads();
#pragma unroll
    for (int kk = 0; kk < BK; kk += 4) {
      v2f af[2], bf[4];
#pragma unroll
      for (int mi = 0; mi < 2; ++mi)
        af[mi] = *(const v2f*)&As[(rt0 + mi * 16 + l16) * LDA + kk + 2 * half];
#pragma unroll
      for (int ni = 0; ni < 4; ++ni)
        bf[ni] = *(const v2f*)&Bst[(ct0 + ni * 16 + l16) * LDB + kk + 2 * half];
#pragma unroll
      for (int mi = 0; mi < 2; ++mi)
#pragma unroll
        for (int ni = 0; ni < 4; ++ni)
          acc[mi][ni] = __builtin_amdgcn_wmma_f32_16x16x4_f32(
              false, af[mi], false, bf[ni], (short)0, acc[mi][ni], false, false);
    }
  }
  // epilogue: D layout -> VGPR r holds (M = r + 8*half, N = l16)
#pragma unroll
  for (int mi = 0; mi < 2; ++mi) {
#pragma unroll
    for (int ni = 0; ni < 4; ++ni) {
      int   col = n0 + ct0 + ni * 16 + l16;
      float bv  = bias[col];
#pragma unroll
      for (int r = 0; r < 8; ++r) {
        int row = m0 + rt0 + mi * 16 + r + 8 * half;
        out[(size_t)row * EMB + col] = acc[mi][ni][r] + bv;
      }
    }
  }
}

// ---------------------------------------------------------------------------
// Flash attention: grid (S/128, H, B). Each wave: 16 queries x DH=64 ctx.
// K tile staged with one async b128/thread; V transposed via async b32.
// ---------------------------------------------------------------------------
#define QT  128
#define KT  16
#define LDK 68   // Ks[key][d] row stride: 16B-aligned rows, bank = 4*lane + c
#define LDV 18   // Vt[d][key]
#define LDP 18   // Pw[q][k]

__global__ __launch_bounds__(256) void attention_kernel(const float* __restrict__ Q,
                                                        const float* __restrict__ K,
                                                        const float* __restrict__ V,
                                                        float* __restrict__ ctx) {
  __shared__ float Ks[KT * LDK];
  __shared__ float Vt[DH * LDV];
  __shared__ float Ps[8 * 16 * LDP];
  const int t    = threadIdx.x;
  const int lane = t & 31;
  const int w    = t >> 5;
  const int half = lane >> 4;
  const int l16  = lane & 15;
  const int b    = blockIdx.z;
  const int h    = blockIdx.y;
  const int q0   = blockIdx.x * QT + w * 16;

  const unsigned long long Kbase = (unsigned long long)(uintptr_t)K;
  const unsigned long long Vbase = (unsigned long long)(uintptr_t)V;

  // Preload Q fragments (A-matrix 16x4 chunks) for this wave's 16 queries
  const float* Qp = Q + ((size_t)(b * SEQ) + q0 + l16) * EMB + h * DH;
  v2f qf[16];
#pragma unroll
  for (int kc = 0; kc < 16; ++kc)
    qf[kc] = *(const v2f*)&Qp[kc * 4 + 2 * half];

  v8f zero = {};
  v8f acc[4];
  acc[0] = zero; acc[1] = zero; acc[2] = zero; acc[3] = zero;
  float mrow[8], lrow[8];
#pragma unroll
  for (int r = 0; r < 8; ++r) { mrow[r] = -3.0e38f; lrow[r] = 0.0f; }

  float* Pw = &Ps[w * 16 * LDP];

  for (int k0 = 0; k0 < SEQ; k0 += KT) {
    __syncthreads();                 // prior-iter LDS reads done
    {
      // K tile: 16x64 floats = 256 float4 -> one async b128 per thread
      int key = t >> 4;
      int d4  = (t & 15) << 2;
      async_g2l_b128(lds_addr_of(&Ks[key * LDK + d4]),
                     (unsigned)((((b * SEQ) + k0 + key) * EMB + h * DH + d4) * 4),
                     Kbase);
      // V tile transposed: 4 async b32 per thread
#pragma unroll
      for (int j = 0; j < 4; ++j) {
        int i   = j * 256 + t;
        int key2 = i >> 6;
        int d    = i & 63;
        async_g2l_b32(lds_addr_of(&Vt[d * LDV + key2]),
                      (unsigned)((((b * SEQ) + k0 + key2) * EMB + h * DH + d) * 4),
                      Vbase);
      }
    }
    wait_async0();
    __syncthreads();

    // S = Q @ K^T : B-frag b0,b1 = K[key=l16][d, d+1] contiguous in Ks
    v8f s = zero;
#pragma unroll
    for (int kc = 0; kc < 16; ++kc) {
      v2f bf = *(const v2f*)&Ks[l16 * LDK + kc * 4 + 2 * half];
      s = __builtin_amdgcn_wmma_f32_16x16x4_f32(false, qf[kc], false, bf,
                                                (short)0, s, false, false);
    }

    // Online softmax: row q = r + 8*half lives in the 16 lanes of this half
#pragma unroll
    for (int r = 0; r < 8; ++r) {
      float sv = s[r] * 0.125f;      // 1/sqrt(64)
      float mx = sv;
      mx = fmaxf(mx, __shfl_xor(mx, 1, 16));
      mx = fmaxf(mx, __shfl_xor(mx, 2, 16));
      mx = fmaxf(mx, __shfl_xor(mx, 4, 16));
      mx = fmaxf(mx, __shfl_xor(mx, 8, 16));
      float mnew  = fmaxf(mrow[r], mx);
      float alpha = __expf(mrow[r] - mnew);
      float p     = __expf(sv - mnew);
      float rs = p;
      rs += __shfl_xor(rs, 1, 16);
      rs += __shfl_xor(rs, 2, 16);
      rs += __shfl_xor(rs, 4, 16);
      rs += __shfl_xor(rs, 8, 16);
      lrow[r] = lrow[r] * alpha + rs;
      mrow[r] = mnew;
#pragma unroll
      for (int dt = 0; dt < 4; ++dt) acc[dt][r] *= alpha;
      Pw[(r + 8 * half) * LDP + l16] = p;   // D-frag -> LDS for A-frag reload
    }
    __syncthreads();

    // ctx += P @ V : A-frag from Pw, B-frag b0,b1 = V[k][d] = Vt[d][k,k+1]
#pragma unroll
    for (int dt = 0; dt < 4; ++dt) {
#pragma unroll
      for (int kc = 0; kc < 4; ++kc) {
        v2f af = *(const v2f*)&Pw[l16 * LDP + kc * 4 + 2 * half];
        v2f bf = *(const v2f*)&Vt[(dt * 16 + l16) * LDV + kc * 4 + 2 * half];
        acc[dt] = __builtin_amdgcn_wmma_f32_16x16x4_f32(false, af, false, bf,
                                                        (short)0, acc[dt], false, false);
      }
    }
  }

  // Finalize: divide by running sum, write concat layout [B,S,H*DH]
#pragma unroll
  for (int dt = 0; dt < 4; ++dt) {
#pragma unroll
    for (int r = 0; r < 8; ++r) {
      int qrow = q0 + r + 8 * half;
      int col  = h * DH + dt * 16 + l16;
      ctx[((size_t)(b * SEQ) + qrow) * EMB + col] = acc[dt][r] / lrow[r];
    }
  }
}

// ---------------------------------------------------------------------------
extern "C" void kernel_launch(void* const* d_in, const int* in_sizes, int n_in,
                              void* d_out, int out_size, void* d_ws, size_t ws_size,
                              hipStream_t stream) {
  const float* x  = (const float*)d_in[0];
  const float* Wq = (const float*)d_in[1];
  const float* bq = (const float*)d_in[2];
  const float* Wk = (const float*)d_in[3];
  const float* bk = (const float*)d_in[4];
  const float* Wv = (const float*)d_in[5];
  const float* bv = (const float*)d_in[6];
  const float* Wo = (const float*)d_in[7];
  const float* bo = (const float*)d_in[8];
  float*       out = (float*)d_out;

  size_t per = (size_t)NTOK * EMB;
  float* Qb = (float*)d_ws;
  float* Kb = Qb + per;
  float* Vb = Kb + per;
  float* Cb = Vb + per;

  dim3 gg(EMB / BN, NTOK / BM);        // (8, 64)
  gemm_bias_kernel<<<gg, 256, 0, stream>>>(x, Wq, bq, Qb);
  gemm_bias_kernel<<<gg, 256, 0, stream>>>(x, Wk, bk, Kb);
  gemm_bias_kernel<<<gg, 256, 0, stream>>>(x, Wv, bv, Vb);

  dim3 ga(SEQ / QT, HEADS, BATCH);     // (16, 16, 4)
  attention_kernel<<<ga, 256, 0, stream>>>(Qb, Kb, Vb, Cb);

  gemm_bias_kernel<<<gg, 256, 0, stream>>>(Cb, Wo, bo, out);
}